// QReLU_71451075936272
// MI455X (gfx1250) — compile-verified
//
#include <hip/hip_runtime.h>

// ReLU over 2^28 f32: HBM-streaming kernel, floor = 2 GiB / 23.3 TB/s ~= 92 us.
// Main kernel: branch-free straight line, 8x global_load_b128 (NT, one clause,
// scale_offset addressing) -> integer-domain ReLU -> 8x global_store_b128 (NT).
//
// ReLU is computed in the integer domain:
//   relu(x) = bit_cast<float>( smax( bit_cast<int>(x), 0 ) )
// This is exact for all non-NaN f32 (incl. -0.0 -> +0.0) and cannot be
// re-folded by the compiler into the float maxnum path, which was observed
// lowering to an ambiguous max(x, x) with no visible zero operand.

typedef __attribute__((ext_vector_type(4))) float v4f;

#define THREADS 256
#define UNROLL  8
#define TILE_V4 (THREADS * UNROLL)   // 2048 vec4 = 32 KiB per block

static __device__ __forceinline__ float relu1(float x) {
  int xi = __float_as_int(x);
  xi = (xi > 0) ? xi : 0;            // lowers to v_max_i32 with literal 0
  return __int_as_float(xi);
}

__global__ __launch_bounds__(THREADS) void relu_main(const v4f* __restrict__ in,
                                                     v4f* __restrict__ out) {
  const unsigned tid = threadIdx.x;
  const size_t base = (size_t)blockIdx.x * TILE_V4;  // scalar (SALU) part
  const v4f* __restrict__ pin  = in  + base;
  v4f* __restrict__       pout = out + base;

  v4f v[UNROLL];
  // All 8 b128 NT loads issued back-to-back (single clause, 8 outstanding
  // LOADcnt = 128 B/lane in flight) before any wait.
#pragma unroll
  for (int u = 0; u < UNROLL; ++u) {
    v[u] = __builtin_nontemporal_load(&pin[tid + u * THREADS]);
  }
#pragma unroll
  for (int u = 0; u < UNROLL; ++u) {
    v4f r;
    r[0] = relu1(v[u][0]);
    r[1] = relu1(v[u][1]);
    r[2] = relu1(v[u][2]);
    r[3] = relu1(v[u][3]);
    __builtin_nontemporal_store(r, &pout[tid + u * THREADS]);
  }
}

// Scalar tail: covers floats [start, n). Never launched for the reference
// shape (2^28 floats tile exactly into 32768 blocks).
__global__ __launch_bounds__(THREADS) void relu_tail(const float* __restrict__ in,
                                                     float* __restrict__ out,
                                                     long long start,
                                                     long long n) {
  long long i = start + (long long)blockIdx.x * THREADS + threadIdx.x;
  if (i < n) {
    out[i] = relu1(in[i]);
  }
}

extern "C" void kernel_launch(void* const* d_in, const int* in_sizes, int n_in,
                              void* d_out, int out_size, void* d_ws, size_t ws_size,
                              hipStream_t stream) {
  (void)n_in; (void)d_ws; (void)ws_size; (void)out_size;

  const float* in = (const float*)d_in[0];
  float* out = (float*)d_out;
  const long long n  = (long long)in_sizes[0];  // 268,435,456 for reference shape
  const long long n4 = n >> 2;                  // whole vec4 count

  const long long fullTiles = n4 / TILE_V4;     // 32,768 for reference
  if (fullTiles > 0) {
    relu_main<<<dim3((unsigned)fullTiles), dim3(THREADS), 0, stream>>>(
        (const v4f*)in, (v4f*)out);
  }

  const long long covered = fullTiles * (long long)TILE_V4 * 4;  // floats done
  const long long tail = n - covered;
  if (tail > 0) {
    const int blocks = (int)((tail + THREADS - 1) / THREADS);
    relu_tail<<<dim3(blocks), dim3(THREADS), 0, stream>>>(in, out, covered, n);
  }
}